// Voxelizer_81028853006683
// MI455X (gfx1250) — compile-verified
//
#include <hip/hip_runtime.h>

// ---- problem constants (from the reference) ----
// X:[0,72) Y:[-16,16) Z:[-2,7.6), voxel 0.2 -> NX=360, NY=160, NZ=48
constexpr int   NXc = 360;
constexpr int   NYc = 160;
constexpr int   NZc = 48;
constexpr int   Cc  = 4;
constexpr int   Bc  = 8;
constexpr int   Sc  = NZc * NYc * NXc;   // 2,764,800 voxels per batch
constexpr int   Gc  = Bc * Sc;           // 22,118,400 voxels total
// 256 threads x 4 voxels = 1024 voxels per block; Sc = 2700 * 1024 exactly,
// so every block lies entirely inside one batch plane.
constexpr int   VOX_PER_BLK    = 1024;
constexpr int   BLKS_PER_BATCH = Sc / VOX_PER_BLK;   // 2700
constexpr int   GATHER_BLKS    = Bc * BLKS_PER_BATCH; // 21600 (== Gc/1024 exactly)
constexpr float X_MINf = 0.0f,  Y_MINf = -16.0f, Z_MINf = -2.0f;
constexpr float VXf = 0.2f, VYf = 0.2f, VZf = 0.2f;

typedef float f32x4 __attribute__((ext_vector_type(4)));
typedef int   i32x4 __attribute__((ext_vector_type(4)));

// ---------------------------------------------------------------------------
// Kernel 1: fill winner[] with sentinel n. Exact grid -> no bounds check.
// Regular (RT) stores on purpose: the 88.5 MB winner array fits in the 192 MB
// global L2, and we want the subsequent atomics + readback to hit L2.
// ---------------------------------------------------------------------------
__global__ void vox_init_winner(int* __restrict__ winner, int sentinel) {
    int t = blockIdx.x * blockDim.x + threadIdx.x;   // one thread : 4 voxels
    i32x4 v = { sentinel, sentinel, sentinel, sentinel };
    *reinterpret_cast<i32x4*>(winner + 4 * (long)t) = v;   // global_store_b128
}

// ---------------------------------------------------------------------------
// Kernel 2: per-point scatter. b128 coalesced point load, voxelize exactly
// like the reference (division + floorf), then a NON-RETURNING
// global_atomic_min into the L2-resident winner array.
// ---------------------------------------------------------------------------
__global__ void vox_scatter_min(const float* __restrict__ rdr_sparse,
                                const int*   __restrict__ batch_indices,
                                int*         __restrict__ winner,
                                int n) {
    int i = blockIdx.x * blockDim.x + threadIdx.x;
    if (i >= n) return;

    f32x4 p = *reinterpret_cast<const f32x4*>(rdr_sparse + 4 * (long)i); // b128 load
    int xi = (int)floorf((p.x - X_MINf) / VXf);
    int yi = (int)floorf((p.y - Y_MINf) / VYf);
    int zi = (int)floorf((p.z - Z_MINf) / VZf);

    bool valid = (xi >= 0) & (xi < NXc) &
                 (yi >= 0) & (yi < NYc) &
                 (zi >= 0) & (zi < NZc);
    if (!valid) return;

    int b    = batch_indices[i];
    int flat = ((b * NZc + zi) * NYc + yi) * NXc + xi;
    atomicMin(winner + flat, i);   // non-returning -> global_atomic_min (no RTN)
}

// ---------------------------------------------------------------------------
// Kernel 3: dense gather + NCZYX transpose. One thread handles 4 consecutive
// voxels; one block handles 1024 voxels inside a single batch plane, so the
// batch/spatial decomposition is pure SALU:
//   - NT b128 load of 4 winner indices          (last use -> don't retain)
//   - up to 4 random NT b128 feature gathers    (each row read at most once)
//   - 4 channel-plane b128 stores with TH_STORE_NT so the 354 MB output
//     stream does not evict L2-resident data.
// Exact grid (21600 blocks) -> no bounds check.
// ---------------------------------------------------------------------------
__global__ void vox_gather_dense(const int*   __restrict__ winner,
                                 const float* __restrict__ rdr_sparse,
                                 float*       __restrict__ out,
                                 int n) {
    int b      = blockIdx.x / BLKS_PER_BATCH;                       // SALU
    int spBase = (blockIdx.x - b * BLKS_PER_BATCH) * VOX_PER_BLK;   // SALU
    int sp     = spBase + (int)threadIdx.x * 4;                     // lane offset
    long g0    = (long)b * Sc + sp;                                 // flat voxel

    i32x4 w = __builtin_nontemporal_load(
        reinterpret_cast<const i32x4*>(winner + g0));   // global_load_b128 NT

    f32x4 f[4];
#pragma unroll
    for (int k = 0; k < 4; ++k) {
        int wi = w[k];
        if (wi < n) {
            f[k] = __builtin_nontemporal_load(
                reinterpret_cast<const f32x4*>(rdr_sparse + 4 * (long)wi));
        } else {
            f[k] = (f32x4){0.0f, 0.0f, 0.0f, 0.0f};
        }
    }

    float* base = out + (long)b * (Cc * Sc) + sp;   // SGPR base + VGPR offset
#pragma unroll
    for (int c = 0; c < Cc; ++c) {
        f32x4 v = { f[0][c], f[1][c], f[2][c], f[3][c] };
        // global_store_b128 ... th:TH_STORE_NT
        __builtin_nontemporal_store(v, reinterpret_cast<f32x4*>(base + (long)c * Sc));
    }
}

// ---------------------------------------------------------------------------
extern "C" void kernel_launch(void* const* d_in, const int* in_sizes, int n_in,
                              void* d_out, int out_size, void* d_ws, size_t ws_size,
                              hipStream_t stream) {
    const float* rdr_sparse    = (const float*)d_in[0];   // [N,4] f32
    const int*   batch_indices = (const int*)  d_in[1];   // [N] int
    // d_in[2] = batch_size scalar (== Bc, baked into constants like NX/NY/NZ)

    const int n = in_sizes[0] / 4;          // number of points
    int* winner = (int*)d_ws;               // Gc * 4 bytes = 88.5 MB scratch
    (void)ws_size; (void)n_in; (void)out_size;

    const int  threads = 256;               // 8 wave32 waves per block
    const dim3 blk(threads);

    vox_init_winner <<<Gc / VOX_PER_BLK, blk, 0, stream>>>(winner, n);      // 21600
    vox_scatter_min <<<(n + threads - 1) / threads, blk, 0, stream>>>(
        rdr_sparse, batch_indices, winner, n);
    vox_gather_dense<<<GATHER_BLKS, blk, 0, stream>>>(
        winner, rdr_sparse, (float*)d_out, n);
}